// RoiPatchBridge_4793183502574
// MI455X (gfx1250) — compile-verified
//
#include <hip/hip_runtime.h>

#define B_    32
#define C_    256
#define H_    112
#define W_    112
#define ROI_  7
#define K_    12544   // C_*ROI_*ROI_
#define HID_  1152
#define MAXG  4
#define SCALE_ 0.25f
#define KSPLIT 8      // split-K factor for HBM saturation

typedef __attribute__((ext_vector_type(2))) float v2f;
typedef __attribute__((ext_vector_type(8))) float v8f;

// ---------------------------------------------------------------------------
// Kernel 1: ROI-align (torchvision semantics, aligned=True, adaptive grid
// capped at MAX_GRID via masking, exactly mirroring the reference math).
// One block per (channel, box); lanes 0..48 each own one 7x7 bin.
// ---------------------------------------------------------------------------
__global__ __launch_bounds__(64)
void roi_align_kernel(const float* __restrict__ feats,
                      const float* __restrict__ boxes,
                      float* __restrict__ pooled) {
  const int c = blockIdx.x;
  const int b = blockIdx.y;
  const int t = threadIdx.x;
  if (t >= ROI_ * ROI_) return;
  const int py = t / ROI_, px = t % ROI_;

  const float* box = boxes + b * 4;
  const float sx = box[0] * SCALE_ - 0.5f;
  const float sy = box[1] * SCALE_ - 0.5f;
  const float ex = box[2] * SCALE_ - 0.5f;
  const float ey = box[3] * SCALE_ - 0.5f;
  const float roi_w = ex - sx, roi_h = ey - sy;
  const float bin_w = roi_w / (float)ROI_, bin_h = roi_h / (float)ROI_;
  const float gw = fminf(fmaxf(ceilf(roi_w / (float)ROI_), 1.0f), (float)MAXG);
  const float gh = fminf(fmaxf(ceilf(roi_h / (float)ROI_), 1.0f), (float)MAXG);

  const float* f = feats + (size_t)(b * C_ + c) * (H_ * W_);
  float acc = 0.0f;
#pragma unroll
  for (int gy = 0; gy < MAXG; ++gy) {
    const float y  = sy + (float)py * bin_h + ((float)gy + 0.5f) * bin_h / gh;
    const bool  vy = (y >= -1.0f) && (y <= (float)H_) && ((float)gy < gh);
    const float yc = fminf(fmaxf(y, 0.0f), (float)(H_ - 1));
    const int   y0 = (int)fminf(floorf(yc), (float)(H_ - 2));
    const float ly = yc - (float)y0, hy = 1.0f - ly;
#pragma unroll
    for (int gx = 0; gx < MAXG; ++gx) {
      const float x  = sx + (float)px * bin_w + ((float)gx + 0.5f) * bin_w / gw;
      const bool  vx = (x >= -1.0f) && (x <= (float)W_) && ((float)gx < gw);
      const float xc = fminf(fmaxf(x, 0.0f), (float)(W_ - 1));
      const int   x0 = (int)fminf(floorf(xc), (float)(W_ - 2));
      const float lx = xc - (float)x0, hx = 1.0f - lx;
      if (vy && vx) {
        const float* r0 = f + y0 * W_ + x0;
        const float* r1 = r0 + W_;
        acc += r0[0] * hy * hx + r0[1] * hy * lx
             + r1[0] * ly * hx + r1[1] * ly * lx;
      }
    }
  }
  pooled[(size_t)b * K_ + c * (ROI_ * ROI_) + t] = acc / (gh * gw);
}

// ---------------------------------------------------------------------------
// Kernel 2: split-K GEMM partials via V_WMMA_F32_16X16X4_F32.
// Grid = (72 N-tiles, 2 M-tiles, 8 K-slices) = 1152 blocks so the chip has
// enough waves in flight to saturate HBM while streaming the 57.8 MB W_proj.
// Block = 128 threads (4 waves); each wave owns K_/(8*4) = 392 K-elements
// (98 WMMA steps of K=4), then an LDS tree reduce; wave 0 writes the partial
// 16x16 tile to partials[kslice].
// A layout: lanes0-15 K={k0,k0+1}, lanes16-31 K={k0+2,k0+3} (float2 load).
// B layout: VGPR v holds row K=k0+2*half+v, N = n0 + (lane&15).
// C layout: VGPR i -> row m0 + i + 8*half, col n0 + (lane&15).
// ---------------------------------------------------------------------------
__global__ __launch_bounds__(128)
void gemm_wmma_splitk(const float* __restrict__ A,        // [32, K_]
                      const float* __restrict__ Wm,       // [K_, HID_]
                      float* __restrict__ partials) {     // [KSPLIT, 32, HID_]
  const int n0   = blockIdx.x * 16;           // 72 tiles
  const int m0   = blockIdx.y * 16;           // 2 tiles
  const int ks   = blockIdx.z;                // 0..KSPLIT-1
  const int wave = threadIdx.x >> 5;          // 0..3 (wave32)
  const int lane = threadIdx.x & 31;
  const int half = lane >> 4;
  const int l16  = lane & 15;

  const float* Arow = A + (size_t)(m0 + l16) * K_;
  v8f c = {};
  int k = ks * (K_ / KSPLIT) + wave * (K_ / KSPLIT / 4);
#pragma unroll 4
  for (int s = 0; s < (K_ / KSPLIT / 16); ++s, k += 4) {   // 98 steps
    const int ka = k + 2 * half;
    const v2f a = *(const v2f*)(Arow + ka);   // A[m][ka], A[m][ka+1]
    const float* wp = Wm + (size_t)ka * HID_ + n0 + l16;
    v2f bb;
    bb.x = wp[0];                             // row ka
    bb.y = wp[HID_];                          // row ka+1
    c = __builtin_amdgcn_wmma_f32_16x16x4_f32(
        /*neg_a=*/false, a, /*neg_b=*/false, bb,
        /*c_mod=*/(short)0, c, /*reuse_a=*/false, /*reuse_b=*/false);
  }

  __shared__ float red[4 * 256];
  float* mine = red + wave * 256 + lane * 8;
#pragma unroll
  for (int i = 0; i < 8; ++i) mine[i] = c[i];
  __syncthreads();

  if (wave == 0) {
    float* pslice = partials + (size_t)ks * (B_ * HID_);
#pragma unroll
    for (int i = 0; i < 8; ++i) {
      const int idx = lane * 8 + i;
      const float s = red[idx] + red[256 + idx] + red[512 + idx] + red[768 + idx];
      const int row = m0 + i + 8 * half;
      const int col = n0 + l16;
      pslice[(size_t)row * HID_ + col] = s;
    }
  }
}

// ---------------------------------------------------------------------------
// Kernel 2b: deterministic split-K reduction (fixed summation order) + bias.
// ---------------------------------------------------------------------------
__global__ __launch_bounds__(256)
void splitk_reduce_kernel(const float* __restrict__ partials, // [KSPLIT,32,HID_]
                          const float* __restrict__ bias,     // [HID_]
                          float* __restrict__ Hout) {         // [32, HID_]
  const int i = blockIdx.x * 256 + threadIdx.x;   // over B_*HID_
  if (i >= B_ * HID_) return;
  float s = bias[i % HID_];
#pragma unroll
  for (int p = 0; p < KSPLIT; ++p) s += partials[(size_t)p * (B_ * HID_) + i];
  Hout[i] = s;
}

// ---------------------------------------------------------------------------
// Kernel 3: LayerNorm over HID_ per batch row + assemble [B, 2, HID_] output.
// ---------------------------------------------------------------------------
__global__ __launch_bounds__(256)
void ln_out_kernel(const float* __restrict__ Hin,
                   const float* __restrict__ gamma,
                   const float* __restrict__ beta,
                   const float* __restrict__ cls,
                   float* __restrict__ out) {
  const int b = blockIdx.x;
  const int t = threadIdx.x;
  __shared__ float s1[256], s2[256];
  float a1 = 0.0f, a2 = 0.0f;
  for (int j = t; j < HID_; j += 256) {
    const float v = Hin[b * HID_ + j];
    a1 += v; a2 += v * v;
  }
  s1[t] = a1; s2[t] = a2;
  __syncthreads();
  for (int o = 128; o > 0; o >>= 1) {
    if (t < o) { s1[t] += s1[t + o]; s2[t] += s2[t + o]; }
    __syncthreads();
  }
  const float mu  = s1[0] * (1.0f / (float)HID_);
  const float var = s2[0] * (1.0f / (float)HID_) - mu * mu;
  const float inv = rsqrtf(var + 1e-5f);

  float* ob = out + (size_t)b * 2 * HID_;
  for (int j = t; j < HID_; j += 256) {
    ob[j] = cls[j];                                   // cls token row
    const float v = Hin[b * HID_ + j];
    ob[HID_ + j] = gamma[j] * (v - mu) * inv + beta[j];
  }
}

// ---------------------------------------------------------------------------
extern "C" void kernel_launch(void* const* d_in, const int* in_sizes, int n_in,
                              void* d_out, int out_size, void* d_ws, size_t ws_size,
                              hipStream_t stream) {
  const float* feats = (const float*)d_in[0];  // [32,256,112,112]
  const float* boxes = (const float*)d_in[1];  // [32,4]
  const float* Wp    = (const float*)d_in[2];  // [12544,1152]
  const float* bp    = (const float*)d_in[3];  // [1152]
  const float* gamma = (const float*)d_in[4];  // [1152]
  const float* beta  = (const float*)d_in[5];  // [1152]
  const float* cls   = (const float*)d_in[6];  // [1,1,1152]
  float* out = (float*)d_out;                  // [32,2,1152]

  float* pooled   = (float*)d_ws;                           // [32, K_]
  float* partials = pooled + (size_t)B_ * K_;               // [KSPLIT, 32, HID_]
  float* Hbuf     = partials + (size_t)KSPLIT * B_ * HID_;  // [32, HID_]

  roi_align_kernel<<<dim3(C_, B_), 64, 0, stream>>>(feats, boxes, pooled);
  gemm_wmma_splitk<<<dim3(HID_ / 16, B_ / 16, KSPLIT), 128, 0, stream>>>(pooled, Wp, partials);
  splitk_reduce_kernel<<<(B_ * HID_ + 255) / 256, 256, 0, stream>>>(partials, bp, Hbuf);
  ln_out_kernel<<<B_, 256, 0, stream>>>(Hbuf, gamma, beta, cls, out);
}